// MambaEncoder_2061584302614
// MI455X (gfx1250) — compile-verified
//
#include <hip/hip_runtime.h>
#include <hip/hip_bf16.h>

typedef float v2f __attribute__((ext_vector_type(2)));
typedef float v8f __attribute__((ext_vector_type(8)));

#define BATCH    4
#define L_SEQ    2048
#define D_MODEL  512
#define HIDDEN   256
#define D_INNER  512
#define N_STATE  16
#define ROWS     (BATCH * L_SEQ)   // 8192

// ---------------------------------------------------------------------------
// Register-blocked fp32 WMMA GEMM:
//   C[M,N] = act( A[M,K] (row stride lda) * W[N,K]^T + bias )
// One wave computes a (16*MT) x (16*NT) strip with MT*NT accumulators,
// reusing each A fragment NT times and each B fragment MT times.
// act: 0 = none, 1 = softplus
// ---------------------------------------------------------------------------
template <int MT, int NT>
__global__ __launch_bounds__(128)
void gemm_wmma_f32(const float* __restrict__ A, const float* __restrict__ W,
                   const float* __restrict__ bias, float* __restrict__ C,
                   int M, int N, int K, int lda, int act)
{
    const int lane    = threadIdx.x & 31;
    const int wave    = blockIdx.x * (blockDim.x >> 5) + (threadIdx.x >> 5);
    const int stripsN = N / (16 * NT);
    const int total   = (M / (16 * MT)) * stripsN;
    if (wave >= total) return;               // wave-uniform: EXEC stays all-ones

    const int tm   = (wave / stripsN) * 16 * MT;
    const int tn   = (wave % stripsN) * 16 * NT;
    const int half = lane >> 4;              // 0: lanes 0-15, 1: lanes 16-31
    const int l16  = lane & 15;

    // A frag: lanes 0-15 hold K=k,k+1 of row M=lane; lanes 16-31 hold K=k+2,k+3
    const float* arow[MT];
#pragma unroll
    for (int i = 0; i < MT; ++i)
        arow[i] = A + (long)(tm + i * 16 + l16) * lda + (half << 1);
    // B frag (B[k][n] = W[n][k]): mirrored striping over K halves
    const float* wrow[NT];
#pragma unroll
    for (int j = 0; j < NT; ++j)
        wrow[j] = W + (long)(tn + j * 16 + l16) * K + (half << 1);

    v8f acc[MT][NT] = {};                    // value-init -> zeros
    for (int k = 0; k < K; k += 4) {
        v2f a[MT], b[NT];
#pragma unroll
        for (int i = 0; i < MT; ++i) a[i] = *(const v2f*)(arow[i] + k);
#pragma unroll
        for (int j = 0; j < NT; ++j) b[j] = *(const v2f*)(wrow[j] + k);
#pragma unroll
        for (int i = 0; i < MT; ++i)
#pragma unroll
            for (int j = 0; j < NT; ++j)
                acc[i][j] = __builtin_amdgcn_wmma_f32_16x16x4_f32(
                    false, a[i], false, b[j], (short)0, acc[i][j], false, false);
    }

#pragma unroll
    for (int j = 0; j < NT; ++j) {
        const int   n  = tn + j * 16 + l16;
        const float bv = bias ? bias[n] : 0.f;
#pragma unroll
        for (int i = 0; i < MT; ++i)
#pragma unroll
            for (int v = 0; v < 8; ++v) {
                const int m = tm + i * 16 + v + (half << 3);
                float r = acc[i][j][v] + bv;
                if (act == 1) r = (r > 20.f) ? r : log1pf(__expf(r));
                C[(long)m * N + n] = r;
            }
    }
}

// ---------------------------------------------------------------------------
// Depthwise causal conv (k=4) + SiLU. rev=1 runs the conv anti-causally,
// which equals the causal conv on the time-flipped sequence (stored unflipped).
// xi = xz[..., 0:512] with row stride 1024.
// ---------------------------------------------------------------------------
__global__ void conv_silu_kernel(const float* __restrict__ xz,
                                 const float* __restrict__ cw,
                                 const float* __restrict__ cb,
                                 float* __restrict__ u, int rev)
{
    const int idx = blockIdx.x * blockDim.x + threadIdx.x;
    if (idx >= BATCH * L_SEQ * D_INNER) return;
    const int c = idx % D_INNER;
    const int t = (idx / D_INNER) % L_SEQ;
    const int b = idx / (D_INNER * L_SEQ);

    const float* base = xz + (long)b * L_SEQ * (2 * D_INNER) + c;
    float acc = cb[c];
#pragma unroll
    for (int k = 0; k < 4; ++k) {
        const int tt = rev ? (t + 3 - k) : (t - 3 + k);
        if (tt >= 0 && tt < L_SEQ)
            acc += cw[c * 4 + k] * base[(long)tt * (2 * D_INNER)];
    }
    u[idx] = acc / (1.f + __expf(-acc));     // silu
}

// ---------------------------------------------------------------------------
// Selective scan. One thread owns one (batch, channel): 16-wide state in
// registers, serial over L (rev=1 iterates backward = flipped-time scan).
// Per-step B/C (shared by all channels) staged in LDS; global_prefetch_b8
// warms the L2->WGP path a few dependent steps ahead.
// ---------------------------------------------------------------------------
__global__ __launch_bounds__(128)
void scan_kernel(const float* __restrict__ dtf, const float* __restrict__ u,
                 const float* __restrict__ xz, const float* __restrict__ dbc,
                 const float* __restrict__ A_log, const float* __restrict__ Dp,
                 float* __restrict__ ysum, int rev)
{
    __shared__ float sBC[32];                 // [0:16)=B_t  [16:32)=C_t
    const int ch = blockIdx.x * 128 + threadIdx.x;
    const int b  = blockIdx.y;

    float Arow[N_STATE];
#pragma unroll
    for (int n = 0; n < N_STATE; ++n) Arow[n] = -__expf(A_log[ch * N_STATE + n]);
    const float Dv = Dp[ch];

    float h[N_STATE];
#pragma unroll
    for (int n = 0; n < N_STATE; ++n) h[n] = 0.f;

    const int PF = 8;                         // prefetch distance (timesteps)
    float acc = 0.f;
    for (int s = 0; s < L_SEQ; ++s) {
        const int  t   = rev ? (L_SEQ - 1 - s) : s;
        const long row = (long)b * L_SEQ + t;
        if (threadIdx.x < 32) sBC[threadIdx.x] = dbc[row * 48 + 16 + threadIdx.x];
        __syncthreads();

        if (s + PF < L_SEQ) {                 // latency hiding for dependent loads
            const int  tp   = rev ? (L_SEQ - 1 - (s + PF)) : (s + PF);
            const long rowp = (long)b * L_SEQ + tp;
            __builtin_prefetch(dtf + rowp * D_INNER + ch, 0, 1);
            __builtin_prefetch(u   + rowp * D_INNER + ch, 0, 1);
            __builtin_prefetch(xz  + rowp * (2 * D_INNER) + D_INNER + ch, 0, 1);
            if (threadIdx.x < 32)
                __builtin_prefetch(dbc + rowp * 48 + 16 + threadIdx.x, 0, 1);
        }

        const float dt = dtf[row * D_INNER + ch];
        const float uv = u  [row * D_INNER + ch];
        const float zv = xz [row * (2 * D_INNER) + D_INNER + ch];
        const float du = dt * uv;
        float y = 0.f;
#pragma unroll
        for (int n = 0; n < N_STATE; ++n) {
            const float dA = __expf(dt * Arow[n]);
            h[n] = dA * h[n] + du * sBC[n];
            y += h[n] * sBC[16 + n];
        }
        acc += (y + uv * Dv) * (zv / (1.f + __expf(-zv)));
        __syncthreads();
    }
    ysum[b * D_INNER + ch] = acc;
}

// ---------------------------------------------------------------------------
// e[b, d*256+o] = (1/L) * sum_ch ysum[d][b][ch] * out_w[d][o][ch]
// (mean over L commutes with the output projection -> big GEMM eliminated)
// ---------------------------------------------------------------------------
__global__ void proj_e_kernel(const float* __restrict__ ysum,
                              const float* __restrict__ out_w,
                              float* __restrict__ e)
{
    const int idx = blockIdx.x * blockDim.x + threadIdx.x;
    if (idx >= 2 * BATCH * HIDDEN) return;
    const int o = idx % HIDDEN;
    const int b = (idx / HIDDEN) % BATCH;
    const int d = idx / (HIDDEN * BATCH);
    const float* ys = ysum + (d * BATCH + b) * D_INNER;
    const float* wr = out_w + ((long)d * HIDDEN + o) * D_INNER;
    float acc = 0.f;
    for (int c = 0; c < D_INNER; ++c) acc += ys[c] * wr[c];
    e[b * (2 * HIDDEN) + d * HIDDEN + o] = acc * (1.f / (float)L_SEQ);
}

// out[b,p] = op_b[p] + sum_j e[b,j] * op_w[p,j]
__global__ void final_kernel(const float* __restrict__ e,
                             const float* __restrict__ op_w,
                             const float* __restrict__ op_b,
                             float* __restrict__ out)
{
    const int idx = blockIdx.x * blockDim.x + threadIdx.x;
    if (idx >= BATCH * HIDDEN) return;
    const int p = idx % HIDDEN;
    const int b = idx / HIDDEN;
    float acc = op_b[p];
    const float* er = e + b * (2 * HIDDEN);
    const float* wr = op_w + (long)p * (2 * HIDDEN);
    for (int j = 0; j < 2 * HIDDEN; ++j) acc += er[j] * wr[j];
    out[idx] = acc;
}

// ---------------------------------------------------------------------------
extern "C" void kernel_launch(void* const* d_in, const int* in_sizes, int n_in,
                              void* d_out, int out_size, void* d_ws, size_t ws_size,
                              hipStream_t stream)
{
    const float* x      = (const float*)d_in[0];
    const float* ip_w   = (const float*)d_in[1];
    const float* ip_b   = (const float*)d_in[2];
    const float* in_w   = (const float*)d_in[3];
    const float* conv_w = (const float*)d_in[4];
    const float* conv_b = (const float*)d_in[5];
    const float* xproj  = (const float*)d_in[6];
    const float* dt_w   = (const float*)d_in[7];
    const float* dt_b   = (const float*)d_in[8];
    const float* A_log  = (const float*)d_in[9];
    const float* Dp     = (const float*)d_in[10];
    const float* out_w  = (const float*)d_in[11];
    const float* op_w   = (const float*)d_in[12];
    const float* op_b   = (const float*)d_in[13];
    float* out = (float*)d_out;

    // workspace partition (floats)
    float* ws   = (float*)d_ws;
    float* h    = ws;                               // 8192*256
    float* xz   = h    + (long)ROWS * HIDDEN;       // 8192*1024
    float* u    = xz   + (long)ROWS * 2 * D_INNER;  // 8192*512
    float* dbc  = u    + (long)ROWS * D_INNER;      // 8192*48
    float* dtf  = dbc  + (long)ROWS * 48;           // 8192*512
    float* ysum = dtf  + (long)ROWS * D_INNER;      // 2*4*512
    float* e    = ysum + 2 * BATCH * D_INNER;       // 4*512

    auto blocks24 = [](int M, int N) {   // MT=2, NT=4 strips, 4 waves/block
        int waves = (M / 32) * (N / 64);
        return (waves + 3) / 4;
    };

    // 1) h = x @ ip_w.T + ip_b          [8192, 256]
    gemm_wmma_f32<2, 4><<<blocks24(ROWS, HIDDEN), 128, 0, stream>>>(
        x, ip_w, ip_b, h, ROWS, HIDDEN, D_MODEL, D_MODEL, 0);

    for (int d = 0; d < 2; ++d) {
        const float* in_w_d = in_w   + (long)d * 1024 * HIDDEN;
        const float* cw_d   = conv_w + (long)d * D_INNER * 4;
        const float* cb_d   = conv_b + (long)d * D_INNER;
        const float* xp_d   = xproj  + (long)d * 48 * D_INNER;
        const float* dtw_d  = dt_w   + (long)d * D_INNER * 16;
        const float* dtb_d  = dt_b   + (long)d * D_INNER;
        const float* Alog_d = A_log  + (long)d * D_INNER * N_STATE;
        const float* Dp_d   = Dp     + (long)d * D_INNER;

        // 2) xz = h @ in_w[d].T         [8192, 1024]  (time-flip commutes)
        gemm_wmma_f32<2, 4><<<blocks24(ROWS, 2 * D_INNER), 128, 0, stream>>>(
            h, in_w_d, nullptr, xz, ROWS, 2 * D_INNER, HIDDEN, HIDDEN, 0);

        // 3) u = silu(causal/anti-causal depthwise conv(xi) + cb)
        conv_silu_kernel<<<(BATCH * L_SEQ * D_INNER + 255) / 256, 256, 0, stream>>>(
            xz, cw_d, cb_d, u, d);

        // 4) dbc = u @ xproj[d].T       [8192, 48]  (48 = 3 N-tiles -> NT=3)
        {
            int waves = (ROWS / 32) * (48 / 48);    // MT=2, NT=3
            gemm_wmma_f32<2, 3><<<(waves + 3) / 4, 128, 0, stream>>>(
                u, xp_d, nullptr, dbc, ROWS, 48, D_INNER, D_INNER, 0);
        }

        // 5) dtf = softplus(dbc[:, :16] @ dt_w[d].T + dt_b[d])   [8192, 512]
        gemm_wmma_f32<2, 4><<<blocks24(ROWS, D_INNER), 128, 0, stream>>>(
            dbc, dtw_d, dtb_d, dtf, ROWS, D_INNER, 16, 48, 1);

        // 6) selective scan (forward for d=0, backward for d=1) + time-sum
        scan_kernel<<<dim3(D_INNER / 128, BATCH), 128, 0, stream>>>(
            dtf, u, xz, dbc, Alog_d, Dp_d, ysum + (long)d * BATCH * D_INNER, d);
    }

    // 7) output projection of the time-mean (mean commutes with linear)
    proj_e_kernel<<<(2 * BATCH * HIDDEN + 255) / 256, 256, 0, stream>>>(ysum, out_w, e);
    final_kernel<<<(BATCH * HIDDEN + 255) / 256, 256, 0, stream>>>(e, op_w, op_b, out);
}